// GCNLayer_12429635354858
// MI455X (gfx1250) — compile-verified
//
#include <hip/hip_runtime.h>
#include <stdint.h>

#define TILE 256
#define DFEAT 96

typedef __attribute__((address_space(3))) uint8_t lds8_t;

__global__ __launch_bounds__(256) void spmm_zero_kernel(float* __restrict__ out, int n) {
    int i4 = blockIdx.x * blockDim.x + threadIdx.x;
    int stride = gridDim.x * blockDim.x;
    int n4 = n >> 2;
    float4* out4 = (float4*)out;
    float4 z; z.x = z.y = z.z = z.w = 0.0f;
    for (int i = i4; i < n4; i += stride) out4[i] = z;
    for (int i = (n4 << 2) + i4; i < n; i += stride) out[i] = 0.0f;
}

// COO SpMM: out[rows[e],:] += vals[e] * embeds[cols[e],:]
// Double-buffered pipeline: tile i+1's edge metadata streams HBM->LDS via async
// b128 loads (ASYNCcnt, partial-counter wait) while tile i computes
// L2-resident gathers + f32 atomic scatter-adds.
__global__ __launch_bounds__(256) void spmm_coo_kernel(const int*   __restrict__ rows,
                                                       const int*   __restrict__ cols,
                                                       const float* __restrict__ vals,
                                                       const float* __restrict__ embeds,
                                                       float*       __restrict__ out,
                                                       int nE, int tilesPerBlock) {
    __shared__ __align__(16) int   s_rows[2][TILE];
    __shared__ __align__(16) int   s_cols[2][TILE];
    __shared__ __align__(16) float s_vals[2][TILE];

    const int numTiles = (nE + TILE - 1) / TILE;
    const int tile0 = blockIdx.x * tilesPerBlock;
    if (tile0 >= numTiles) return;
    int tileEnd = tile0 + tilesPerBlock;
    if (tileEnd > numTiles) tileEnd = numTiles;

    const int t = threadIdx.x;

    // Issue async staging of tile ti into buffer b.
    // Full tile: exactly 1 b128 async op per thread for t<192 (wave-uniform), 0 else.
    // Tail tile: exactly 3 b32 async ops per thread.
    auto issue = [&](int ti, int b) {
        const int base = ti * TILE;
        uint32_t ldsR = (uint32_t)(uintptr_t)(lds8_t*)&s_rows[b][0];
        uint32_t ldsC = (uint32_t)(uintptr_t)(lds8_t*)&s_cols[b][0];
        uint32_t ldsV = (uint32_t)(uintptr_t)(lds8_t*)&s_vals[b][0];
        if (base + TILE <= nE) {
            if (t < 64) {
                uint32_t voff = (uint32_t)(base + t * 4) * 4u;
                uint32_t l = ldsR + (uint32_t)t * 16u;
                asm volatile("global_load_async_to_lds_b128 %0, %1, %2"
                             :: "v"(l), "v"(voff), "s"(rows) : "memory");
            } else if (t < 128) {
                int c = t - 64;
                uint32_t voff = (uint32_t)(base + c * 4) * 4u;
                uint32_t l = ldsC + (uint32_t)c * 16u;
                asm volatile("global_load_async_to_lds_b128 %0, %1, %2"
                             :: "v"(l), "v"(voff), "s"(cols) : "memory");
            } else if (t < 192) {
                int c = t - 128;
                uint32_t voff = (uint32_t)(base + c * 4) * 4u;
                uint32_t l = ldsV + (uint32_t)c * 16u;
                asm volatile("global_load_async_to_lds_b128 %0, %1, %2"
                             :: "v"(l), "v"(voff), "s"(vals) : "memory");
            }
        } else {
            int g = base + t;
            if (g >= nE) g = nE - 1;            // clamp; slots >= count are never read
            uint32_t voff = (uint32_t)g * 4u;
            asm volatile("global_load_async_to_lds_b32 %0, %1, %2"
                         :: "v"(ldsR + (uint32_t)t * 4u), "v"(voff), "s"(rows) : "memory");
            asm volatile("global_load_async_to_lds_b32 %0, %1, %2"
                         :: "v"(ldsC + (uint32_t)t * 4u), "v"(voff), "s"(cols) : "memory");
            asm volatile("global_load_async_to_lds_b32 %0, %1, %2"
                         :: "v"(ldsV + (uint32_t)t * 4u), "v"(voff), "s"(vals) : "memory");
        }
    };

    const int lane = t & 31;
    const int wave = t >> 5;

    issue(tile0, tile0 & 1);                    // prologue

    for (int ti = tile0; ti < tileEnd; ++ti) {
        const int b = ti & 1;
        if (ti + 1 < tileEnd) {
            issue(ti + 1, (ti + 1) & 1);        // overlap next tile's HBM streaming
            if ((ti + 1) * TILE + TILE <= nE)
                asm volatile("s_wait_asynccnt 1" ::: "memory");   // drain prev, keep new in flight
            else
                asm volatile("s_wait_asynccnt 3" ::: "memory");   // tail issued 3 ops/wave
        } else {
            asm volatile("s_wait_asynccnt 0" ::: "memory");
        }
        __syncthreads();                        // all waves' staging of tile ti visible

        const int base = ti * TILE;
        int count = nE - base;
        if (count > TILE) count = TILE;

        // One edge per wave-iteration; lane l covers features l, l+32, l+64.
#pragma unroll 2
        for (int e = wave; e < count; e += 8) {
            int   col = s_cols[b][e];           // wave-uniform LDS broadcast
            int   row = s_rows[b][e];
            float v   = s_vals[b][e];

            if (e + 16 < count) {               // warm L2/L0 (global_prefetch_b8)
                int pc = s_cols[b][e + 16];
                __builtin_prefetch(embeds + (size_t)pc * DFEAT, 0, 1);
            }

            const float* src = embeds + (size_t)col * DFEAT + lane;
            float m0 = v * src[0];
            float m1 = v * src[32];
            float m2 = v * src[64];

            float* dst = out + (size_t)row * DFEAT + lane;
            atomicAdd(dst,      m0);
            atomicAdd(dst + 32, m1);
            atomicAdd(dst + 64, m2);
        }
        __syncthreads();                        // buffer b free for re-issue at ti+2
    }
}

extern "C" void kernel_launch(void* const* d_in, const int* in_sizes, int n_in,
                              void* d_out, int out_size, void* d_ws, size_t ws_size,
                              hipStream_t stream) {
    const int*   rows   = (const int*)d_in[0];
    const int*   cols   = (const int*)d_in[1];
    const float* vals   = (const float*)d_in[2];
    const float* embeds = (const float*)d_in[3];
    float*       out    = (float*)d_out;

    const int nE = in_sizes[0];

    // d_out is poisoned; zero it before atomic accumulation (same stream -> ordered).
    spmm_zero_kernel<<<512, 256, 0, stream>>>(out, out_size);

    const int numTiles = (nE + TILE - 1) / TILE;
    const int tilesPerBlock = 4;                // pipeline depth per block
    const int nBlocks = (numTiles + tilesPerBlock - 1) / tilesPerBlock;
    spmm_coo_kernel<<<nBlocks, 256, 0, stream>>>(rows, cols, vals, embeds, out,
                                                 nE, tilesPerBlock);
}